// Linear_DS_agent_29755533427581
// MI455X (gfx1250) — compile-verified
//
#include <hip/hip_runtime.h>
#include <hip/hip_bf16.h>
#include <math.h>

typedef __attribute__((ext_vector_type(2))) float v2f;
typedef __attribute__((ext_vector_type(8))) float v8f;

#define HS 64
#define NT 2048
#define NSTEPS 1000
#define HNN 256
#define TCH 20          // time steps per wave in the WMMA kernel
#define DT 0.01f

// ---------------------------------------------------------------------------
// K1: x0 = (relu(target @ l1_w^T + l1_b) @ l2_w^T + l2_b)^T   -> (64, 2048)
// one block per target, 256 threads
// ---------------------------------------------------------------------------
__global__ __launch_bounds__(256) void mlp_kernel(
    const float* __restrict__ target, const float* __restrict__ l1w,
    const float* __restrict__ l1b, const float* __restrict__ l2w,
    const float* __restrict__ l2b, float* __restrict__ x0) {
  __shared__ float h[HNN];
  const int j = blockIdx.x;
  const int i = threadIdx.x;
  const float t0 = target[j * 2 + 0];
  const float t1 = target[j * 2 + 1];
  float v = fmaf(l1w[i * 2 + 0], t0, fmaf(l1w[i * 2 + 1], t1, l1b[i]));
  h[i] = v > 0.f ? v : 0.f;
  __syncthreads();
  if (i < HS) {
    float acc = l2b[i];
    const float* w = l2w + i * HNN;
#pragma unroll 8
    for (int k = 0; k < HNN; ++k) acc = fmaf(w[k], h[k], acc);
    x0[i * NT + j] = acc;
  }
}

// ---------------------------------------------------------------------------
// K2: invP = inv(P) via Gauss-Jordan with partial pivoting, one block
// ---------------------------------------------------------------------------
__global__ __launch_bounds__(256) void invert_kernel(
    const float* __restrict__ P, float* __restrict__ invP) {
  __shared__ float Aug[HS][2 * HS + 1];  // padded
  __shared__ float fac[HS];
  __shared__ int piv;
  const int tid = threadIdx.x;
  for (int idx = tid; idx < HS * HS; idx += 256) {
    int r = idx / HS, c = idx % HS;
    Aug[r][c] = P[idx];
    Aug[r][HS + c] = (r == c) ? 1.f : 0.f;
  }
  __syncthreads();
  for (int p = 0; p < HS; ++p) {
    if (tid == 0) {
      int best = p;
      float bv = fabsf(Aug[p][p]);
      for (int r = p + 1; r < HS; ++r) {
        float v = fabsf(Aug[r][p]);
        if (v > bv) { bv = v; best = r; }
      }
      piv = best;
    }
    __syncthreads();
    const int pr = piv;
    if (pr != p) {
      for (int c = tid; c < 2 * HS; c += 256) {
        float t = Aug[p][c]; Aug[p][c] = Aug[pr][c]; Aug[pr][c] = t;
      }
    }
    __syncthreads();
    const float inv = 1.0f / Aug[p][p];
    for (int c = tid; c < 2 * HS; c += 256) Aug[p][c] *= inv;
    __syncthreads();
    if (tid < HS) fac[tid] = Aug[tid][p];
    __syncthreads();
    for (int idx = tid; idx < HS * 2 * HS; idx += 256) {
      int r = idx >> 7, c = idx & 127;
      if (r != p) Aug[r][c] = fmaf(-fac[r], Aug[p][c], Aug[r][c]);
    }
    __syncthreads();
  }
  for (int idx = tid; idx < HS * HS; idx += 256)
    invP[idx] = Aug[idx / HS][HS + (idx % HS)];
}

// ---------------------------------------------------------------------------
// K3a: lambda tables. lamv[k] = lam_k ; Lam[t][k] = lam_k^t ; LamT[k][t]
// ---------------------------------------------------------------------------
__global__ __launch_bounds__(64) void lambda_kernel(
    const float* __restrict__ D, float* __restrict__ lamv,
    float* __restrict__ Lam, float* __restrict__ LamT) {
  const int k = threadIdx.x;
  const float lam = 1.0f - DT * __expf(D[k]);
  lamv[k] = lam;
  float v = 1.0f;
  for (int t = 0; t < NSTEPS; ++t) {
    Lam[t * HS + k] = v;
    LamT[k * NSTEPS + t] = v;
    v *= lam;
  }
}

// K3b: WA = lm_w @ P  (2 x 64)
__global__ __launch_bounds__(128) void wa_kernel(
    const float* __restrict__ lmw, const float* __restrict__ P,
    float* __restrict__ WA) {
  const int a = threadIdx.x >> 6;
  const int k = threadIdx.x & 63;
  float acc = 0.f;
  for (int i = 0; i < HS; ++i) acc = fmaf(lmw[a * HS + i], P[i * HS + k], acc);
  WA[a * HS + k] = acc;
}

// K3c: y0 = invP @ x0   (64 x 2048)
__global__ __launch_bounds__(256) void y0_kernel(
    const float* __restrict__ invP, const float* __restrict__ x0,
    float* __restrict__ y0) {
  const int j = blockIdx.x * 256 + threadIdx.x;
  const int i = blockIdx.y;
  float acc = 0.f;
  const float* row = invP + i * HS;
#pragma unroll 8
  for (int k = 0; k < HS; ++k) acc = fmaf(row[k], x0[k * NT + j], acc);
  y0[i * NT + j] = acc;
}

// ---------------------------------------------------------------------------
// K4: actions[j][a][t] = tanh( sum_k WA[a][k] * LamT[k][t] * y0[k][j] + b[a] )
// grid (2048, 4), block 256 over t
// ---------------------------------------------------------------------------
__global__ __launch_bounds__(256) void actions_kernel(
    const float* __restrict__ LamT, const float* __restrict__ y0,
    const float* __restrict__ WA, const float* __restrict__ lmb,
    float* __restrict__ actions) {
  const int j = blockIdx.x;
  const int t = blockIdx.y * 256 + threadIdx.x;
  if (t >= NSTEPS) return;
  float a0 = lmb[0], a1 = lmb[1];
#pragma unroll 4
  for (int k = 0; k < HS; ++k) {
    float z = LamT[k * NSTEPS + t] * y0[k * NT + j];
    a0 = fmaf(WA[k], z, a0);
    a1 = fmaf(WA[HS + k], z, a1);
  }
  actions[(size_t)j * 2 * NSTEPS + t] = tanhf(a0);
  actions[(size_t)j * 2 * NSTEPS + NSTEPS + t] = tanhf(a1);
}

// ---------------------------------------------------------------------------
// K5: hidden[t] = P @ diag(lam^t) @ y0  via V_WMMA_F32_16X16X4_F32
// Block = 64 threads = 2 waves; wave w owns row-tiles {2w, 2w+1} of P and a
// 16-column j-tile for TCH time steps.  A-fragments, y0-fragments, lambda
// fragments and running lambda^t power fragments are ALL register resident;
// the steady-state loop is pure v_pk_mul + v_wmma + stores (no VMEM loads,
// no wait counters on the critical path).
// grid (NT/16 = 128, NSTEPS/TCH = 50), block 64
// ---------------------------------------------------------------------------
__global__ __launch_bounds__(64, 1) void hidden_wmma_kernel(
    const float* __restrict__ P, const float* __restrict__ lamv,
    const float* __restrict__ Lam, const float* __restrict__ y0,
    float* __restrict__ hidden) {
  const int lane = threadIdx.x & 31;
  const int wav = threadIdx.x >> 5;   // 0 or 1
  const int half = lane >> 4;         // K sub-lanes: 0 -> {+0,+1}, 1 -> {+2,+3}
  const int l16 = lane & 15;
  const int j0 = blockIdx.x * 16;
  const int t0 = blockIdx.y * TCH;
  const int rt0 = wav * 2;            // first of the two row-tiles this wave owns

  // A fragments: a[rt][kf] element r = P[(rt0+rt)*16 + l16][4*kf + 2*half + r]
  v2f a[2][16];
#pragma unroll
  for (int rt = 0; rt < 2; ++rt)
#pragma unroll
    for (int kf = 0; kf < 16; ++kf)
      a[rt][kf] =
          *(const v2f*)(P + ((rt0 + rt) * 16 + l16) * HS + kf * 4 + 2 * half);

  // y0 fragments, lambda fragments, and lambda^t0 power fragments
  v2f y[16], lm[16], pw[16];
#pragma unroll
  for (int kf = 0; kf < 16; ++kf) {
    const int kb = 4 * kf + 2 * half;
    y[kf].x = y0[(kb + 0) * NT + j0 + l16];
    y[kf].y = y0[(kb + 1) * NT + j0 + l16];
    lm[kf] = *(const v2f*)(lamv + kb);
    pw[kf] = *(const v2f*)(Lam + t0 * HS + kb);
  }

  for (int tt = 0; tt < TCH; ++tt) {
    v8f c0 = {0.f, 0.f, 0.f, 0.f, 0.f, 0.f, 0.f, 0.f};
    v8f c1 = c0;
#pragma unroll
    for (int kf = 0; kf < 16; ++kf) {
      const v2f b = pw[kf] * y[kf];
      c0 = __builtin_amdgcn_wmma_f32_16x16x4_f32(false, a[0][kf], false, b,
                                                 (short)0, c0, false, false);
      c1 = __builtin_amdgcn_wmma_f32_16x16x4_f32(false, a[1][kf], false, b,
                                                 (short)0, c1, false, false);
      pw[kf] = pw[kf] * lm[kf];  // advance lambda^t in registers
    }
    // C/D layout: reg v, lanes 0-15 -> M=v, lanes 16-31 -> M=v+8 ; N=l16
    float* out =
        hidden + (size_t)(t0 + tt) * HS * NT + (size_t)rt0 * 16 * NT + j0 + l16;
#pragma unroll
    for (int v = 0; v < 8; ++v) {
      const int m = v + 8 * half;
      out[(0 * 16 + m) * NT] = c0[v];
      out[(1 * 16 + m) * NT] = c1[v];
    }
  }
}

// ---------------------------------------------------------------------------
extern "C" void kernel_launch(void* const* d_in, const int* in_sizes, int n_in,
                              void* d_out, int out_size, void* d_ws,
                              size_t ws_size, hipStream_t stream) {
  const float* target = (const float*)d_in[0];
  const float* D      = (const float*)d_in[1];
  const float* P      = (const float*)d_in[2];
  const float* l1w    = (const float*)d_in[3];
  const float* l1b    = (const float*)d_in[4];
  const float* l2w    = (const float*)d_in[5];
  const float* l2b    = (const float*)d_in[6];
  const float* lmw    = (const float*)d_in[7];
  const float* lmb    = (const float*)d_in[8];

  float* out = (float*)d_out;
  float* actions = out;                              // (2048, 2, 1000)
  float* hidden = out + (size_t)NT * 2 * NSTEPS;     // (1000, 64, 2048)

  float* x0   = (float*)d_ws;            // 64*2048
  float* invP = x0 + HS * NT;            // 64*64
  float* lamv = invP + HS * HS;          // 64
  float* Lam  = lamv + HS;               // 1000*64 (t-major)
  float* LamT = Lam + NSTEPS * HS;       // 64*1000 (k-major)
  float* WA   = LamT + HS * NSTEPS;      // 2*64
  float* y0   = WA + 2 * HS;             // 64*2048

  mlp_kernel<<<NT, 256, 0, stream>>>(target, l1w, l1b, l2w, l2b, x0);
  invert_kernel<<<1, 256, 0, stream>>>(P, invP);
  lambda_kernel<<<1, 64, 0, stream>>>(D, lamv, Lam, LamT);
  wa_kernel<<<1, 128, 0, stream>>>(lmw, P, WA);
  y0_kernel<<<dim3(NT / 256, HS), 256, 0, stream>>>(invP, x0, y0);
  actions_kernel<<<dim3(NT, (NSTEPS + 255) / 256), 256, 0, stream>>>(
      LamT, y0, WA, lmb, actions);
  hidden_wmma_kernel<<<dim3(NT / 16, NSTEPS / TCH), 64, 0, stream>>>(
      P, lamv, Lam, y0, hidden);
}